// SpectralConv_Nd_ParaModeScaling_RealVersion_26250840113684
// MI455X (gfx1250) — compile-verified
//
#include <hip/hip_runtime.h>
#include <math.h>

typedef __attribute__((ext_vector_type(2))) float v2f;
typedef __attribute__((ext_vector_type(8))) float v8f;
typedef __attribute__((ext_vector_type(4))) unsigned int u32x4;
typedef __attribute__((ext_vector_type(8))) int i32x8;
typedef __attribute__((ext_vector_type(4))) int i32x4;

// ---------------- WMMA f32 16x16x4 wrapper ----------------
#if defined(__gfx1250__) && __has_builtin(__builtin_amdgcn_wmma_f32_16x16x4_f32)
__device__ __forceinline__ v8f wmma_f32(v2f a, v2f b, v8f c) {
  return __builtin_amdgcn_wmma_f32_16x16x4_f32(false, a, false, b, (short)0, c, false, false);
}
#else
__device__ __forceinline__ v8f wmma_f32(v2f a, v2f b, v8f c) { return c; }
#endif

// ---------------- TDM staging: global -> LDS via Tensor Data Mover ----------------
#if defined(__gfx1250__) && __has_builtin(__builtin_amdgcn_tensor_load_to_lds)
#define HAVE_TDM 1
#endif

// Stage `nfloats` contiguous floats from global into the start of this block's LDS.
// One TDM descriptor issued by wave 0; completion via TENSORcnt; then block barrier.
__device__ __forceinline__ void lds_stage(float* sh, const float* __restrict__ g, int nfloats) {
#ifdef HAVE_TDM
  if ((threadIdx.x >> 5) == 0) {
    unsigned long long ga = (unsigned long long)(size_t)g;
    u32x4 g0;
    g0[0] = 1u;                                   // count=1, user descriptor
    g0[1] = 0u;                                   // lds_addr = base of block LDS
    g0[2] = (unsigned)(ga & 0xffffffffu);         // global_addr[95:64] = low 32
    g0[3] = (unsigned)((ga >> 32) & 0x01ffffffu)  // global_addr[120:96]
            | (2u << 30);                         // type = 2 (image/tensor)
    i32x8 g1;
    g1[0] = 0x00020000;                           // workgroup_mask=0, data_size=2 (4B)
    g1[1] = (int)((unsigned)(nfloats & 0xffff) << 16);        // tensor_dim0[15:0] @63:48
    g1[2] = (int)((((unsigned)nfloats >> 16) & 0xffffu)       // tensor_dim0[31:16]
                  | (1u << 16));                              // tensor_dim1 = 1
    g1[3] = (int)((unsigned)(nfloats & 0xffff) << 16);        // tile_dim0 @127:112
    g1[4] = 1;                                    // tile_dim1 = 1
    g1[5] = nfloats;                              // tensor_dim0_stride low 32
    g1[6] = 0;
    g1[7] = 0;
    i32x4 z4 = (i32x4){0, 0, 0, 0};
#if __clang_major__ >= 23
    i32x8 z8 = (i32x8){0, 0, 0, 0, 0, 0, 0, 0};
    __builtin_amdgcn_tensor_load_to_lds(g0, g1, z4, z4, z8, 0);
#else
    __builtin_amdgcn_tensor_load_to_lds(g0, g1, z4, z4, 0);
#endif
#if __has_builtin(__builtin_amdgcn_s_wait_tensorcnt)
    __builtin_amdgcn_s_wait_tensorcnt(0);
#else
    asm volatile("s_wait_tensorcnt 0x0" ::: "memory");
#endif
  }
  // The TDM write to LDS is invisible to the compiler's alias analysis (it goes
  // through the D#, not through `sh`). Escape the pointer with a memory clobber
  // so subsequent ds_loads from the staged buffer are not folded to undef.
  asm volatile("" : : "v"(sh) : "memory");
#else
  for (int t = threadIdx.x; t < (nfloats >> 2); t += 256)
    ((float4*)sh)[t] = ((const float4*)g)[t];
#endif
  __syncthreads();
}

// ---------------- problem constants ----------------
constexpr int B_ = 16, CI = 32, CO = 32, H = 256, W = 256;
constexpr long NROW = (long)B_ * CI * H;   // 131072 rows for row-DFT GEMMs

// workspace layout (float offsets)
constexpr long WS_MATA = 0;                  // [256][64]  row-DFT  (cos | -sin)/16
constexpr long WS_CB   = 16384;              // [64][256]  cos(kx'*h)/16  ([kx][h])
constexpr long WS_SB   = 32768;              // [64][256]  sin(kx'*h)/16  ([kx][h])
constexpr long WS_CD   = 49152;              // [256][64]  cos(kx'*h)/16  ([h][kx])
constexpr long WS_SD   = 65536;              // [256][64]  sin(kx'*h)/16  ([h][kx])
constexpr long WS_MATE = 81920;              // [64][256]  inv row-DFT (a*cos | -a*sin)/16
constexpr long WS_S    = 98304;              // [16][32][32] hierarchical mode scaling
constexpr long WS_XROW = 114688;             // [131072][64] (re0..31|im0..31); reused as Ycol
constexpr long WS_XFTR = WS_XROW + NROW * 64;
constexpr long WS_XFTI = WS_XFTR + 512L * 2048;
constexpr long WS_OFTR = WS_XFTI + 512L * 2048;
constexpr long WS_OFTI = WS_OFTR + 512L * 2048;

// ---------------- precompute: DFT matrices + mode scaling ----------------
__global__ void __launch_bounds__(256) precompute_kernel(float* __restrict__ ws,
                                                         const float* __restrict__ fs) {
  int idx = blockIdx.x * 256 + threadIdx.x;        // 7 * 16384 entries
  int table = idx >> 14;
  int i = idx & 16383;
  const float inv16 = 1.0f / 16.0f;
  const float twopi = 6.28318530717958647692f;
  if (table == 0) {                 // MatA [w][col]: col<32 -> cos, col>=32 -> -sin
    int w = i >> 6, col = i & 63, ky = col & 31;
    float ang = twopi * (float)((ky * w) & 255) * (1.0f / 256.0f);
    ws[WS_MATA + i] = ((col < 32) ? cosf(ang) : -sinf(ang)) * inv16;
  } else if (table == 1 || table == 2) {  // C_B / S_B [kx][h]
    int kx = i >> 8, h = i & 255;
    int kxp = (kx < 32) ? kx : (192 + kx);
    float ang = twopi * (float)((kxp * h) & 255) * (1.0f / 256.0f);
    ws[(table == 1 ? WS_CB : WS_SB) + i] = (table == 1 ? cosf(ang) : sinf(ang)) * inv16;
  } else if (table == 3 || table == 4) {  // C_D / S_D [h][kx]
    int h = i >> 6, kx = i & 63;
    int kxp = (kx < 32) ? kx : (192 + kx);
    float ang = twopi * (float)((kxp * h) & 255) * (1.0f / 256.0f);
    ws[(table == 3 ? WS_CD : WS_SD) + i] = (table == 3 ? cosf(ang) : sinf(ang)) * inv16;
  } else if (table == 5) {          // MatE [row][w]: row<32 -> a*cos, else -a*sin
    int row = i >> 8, w = i & 255, ky = row & 31;
    float alpha = (ky == 0) ? 1.0f : 2.0f;
    float ang = twopi * (float)((ky * w) & 255) * (1.0f / 256.0f);
    ws[WS_MATE + i] = ((row < 32) ? cosf(ang) : -sinf(ang)) * alpha * inv16;
  } else {                          // s[b][x][y] hierarchical quadrant scaling
    int b = i >> 10, x = (i >> 5) & 31, y = i & 31;
    int mm = (x > y) ? x : y;
    int col;
    if (mm < 8) col = 0;
    else {
      int d = (mm < 16) ? 1 : 2;
      int l0 = 8 << (d - 1);
      if (x < l0) col = 3 * d - 2;
      else if (y < l0) col = 3 * d - 1;
      else col = 3 * d;
    }
    ws[WS_S + i] = fs[b * 7 + col];
  }
}

// ---------------- Stage A: pruned row DFT  [131072x256] x [256x64] ----------------
__global__ void __launch_bounds__(256) stageA_kernel(const float* __restrict__ x,
                                                     const float* __restrict__ Bm,
                                                     float* __restrict__ xrow) {
  __shared__ float shB[16384];                // MatA [256][64], 64 KB
  lds_stage(shB, Bm, 16384);
  int wid = (int)((blockIdx.x * 256 + threadIdx.x) >> 5);
  int l = threadIdx.x & 31, lrow = l & 15, khalf = l >> 4;
  long row = (long)wid * 16 + lrow;
  const float* Ar = x + row * 256;
  v8f a0{}, a1{}, a2{}, a3{};
  for (int k = 0; k < 256; k += 4) {
    int ka = k + 2 * khalf;
    v2f a; a.x = Ar[ka]; a.y = Ar[ka + 1];
    const float* b0 = shB + ka * 64;
    const float* b1 = b0 + 64;
    v2f b;
    b.x = b0[lrow];      b.y = b1[lrow];      a0 = wmma_f32(a, b, a0);
    b.x = b0[16 + lrow]; b.y = b1[16 + lrow]; a1 = wmma_f32(a, b, a1);
    b.x = b0[32 + lrow]; b.y = b1[32 + lrow]; a2 = wmma_f32(a, b, a2);
    b.x = b0[48 + lrow]; b.y = b1[48 + lrow]; a3 = wmma_f32(a, b, a3);
  }
  long m0 = (long)wid * 16;
  for (int j = 0; j < 8; ++j) {
    float* Dr = xrow + (m0 + j + 8 * khalf) * 64;
    Dr[lrow] = a0[j]; Dr[16 + lrow] = a1[j]; Dr[32 + lrow] = a2[j]; Dr[48 + lrow] = a3[j];
  }
}

// ---------------- Stage B: pruned column DFT (complex) ----------------
__global__ void __launch_bounds__(256) stageB_kernel(const float* __restrict__ xrow,
                                                     const float* __restrict__ ws,
                                                     float* __restrict__ xft_re,
                                                     float* __restrict__ xft_im) {
  __shared__ float shCS[32768];               // cos|sin in [h][kx] layout, 128 KB
  lds_stage(shCS, ws + WS_CD, 32768);         // C_D then S_D (contiguous)
  const float* shC = shCS;
  const float* shS = shCS + 16384;
  int wid = (int)((blockIdx.x * 256 + threadIdx.x) >> 5);
  int img = wid >> 3;                 // one image per block (8 waves)
  int tg = wid & 7;
  int mtile = tg >> 1, kyt = tg & 1;
  int l = threadIdx.x & 31, lrow = l & 15, khalf = l >> 4;
  const float* Xr = xrow + (long)img * 256 * 64;            // [h][64]
  int mrow = mtile * 16 + lrow;       // kx row
  int nre = kyt * 16 + lrow;
  v8f Pa{}, Pb{}, Qa{}, Qb{};
  for (int k = 0; k < 256; k += 4) {
    int ka = k + 2 * khalf;           // h index
    v2f ac; ac.x = shC[ka * 64 + mrow]; ac.y = shC[(ka + 1) * 64 + mrow];
    v2f as; as.x = shS[ka * 64 + mrow]; as.y = shS[(ka + 1) * 64 + mrow];
    v2f br; br.x = Xr[(long)ka * 64 + nre];      br.y = Xr[(long)(ka + 1) * 64 + nre];
    v2f bi; bi.x = Xr[(long)ka * 64 + 32 + nre]; bi.y = Xr[(long)(ka + 1) * 64 + 32 + nre];
    Pa = wmma_f32(ac, br, Pa);
    Pb = wmma_f32(ac, bi, Pb);
    Qa = wmma_f32(as, br, Qa);
    Qb = wmma_f32(as, bi, Qb);
  }
  float* dr = xft_re + (long)img * 2048;
  float* di = xft_im + (long)img * 2048;
  for (int j = 0; j < 8; ++j) {
    int kx = mtile * 16 + j + 8 * khalf;
    int ky = kyt * 16 + lrow;
    dr[kx * 32 + ky] = Pa[j] + Qb[j];   // Re = C*a + S*b
    di[kx * 32 + ky] = Pb[j] - Qa[j];   // Im = C*b - S*a
  }
}

// ---------------- Stage C: per-mode complex channel mixing + scaled extra ----------------
__global__ void __launch_bounds__(256) stageC_kernel(const float* __restrict__ xft_re,
                                                     const float* __restrict__ xft_im,
                                                     const float* __restrict__ w0,
                                                     const float* __restrict__ w1,
                                                     const float* __restrict__ we0,
                                                     const float* __restrict__ we1,
                                                     const float* __restrict__ ws_s,
                                                     float* __restrict__ oft_re,
                                                     float* __restrict__ oft_im) {
  int wid = (int)((blockIdx.x * 256 + threadIdx.x) >> 5);
  int otile = wid & 1;
  int my = (wid >> 1) & 31;
  int mx = (wid >> 6) & 31;
  int c = wid >> 11;
  const float* Wb = c ? w1 : w0;
  const float* We = c ? we1 : we0;
  int l = threadIdx.x & 31, lrow = l & 15, khalf = l >> 4;
  int o = otile * 16 + lrow;
  int kxoff = c * 32 + mx;
  v8f t1r{}, t1i{}, t2r{}, t2i{};
  for (int kk = 0; kk < 32; kk += 4) {
    int ka = kk + 2 * khalf;
    long xoff0 = ((long)(lrow * 32 + ka)) * 2048 + kxoff * 32 + my;
    long xoff1 = xoff0 + 2048;
    v2f xr;  xr.x = xft_re[xoff0];  xr.y = xft_re[xoff1];
    v2f xi;  xi.x = xft_im[xoff0];  xi.y = xft_im[xoff1];
    v2f nxi; nxi.x = -xi.x; nxi.y = -xi.y;
    long woff0 = ((((long)ka * 32 + o) * 32 + mx) * 32 + my) * 2;
    long woff1 = woff0 + 32L * 32 * 32 * 2;
    v2f p0 = *(const v2f*)(Wb + woff0);
    v2f p1 = *(const v2f*)(Wb + woff1);
    v2f br; br.x = p0.x; br.y = p1.x;
    v2f bi; bi.x = p0.y; bi.y = p1.y;
    t1r = wmma_f32(xr, br, t1r);  t1r = wmma_f32(nxi, bi, t1r);
    t1i = wmma_f32(xr, bi, t1i);  t1i = wmma_f32(xi, br, t1i);
    v2f q0 = *(const v2f*)(We + woff0);
    v2f q1 = *(const v2f*)(We + woff1);
    br.x = q0.x; br.y = q1.x; bi.x = q0.y; bi.y = q1.y;
    t2r = wmma_f32(xr, br, t2r);  t2r = wmma_f32(nxi, bi, t2r);
    t2i = wmma_f32(xr, bi, t2i);  t2i = wmma_f32(xi, br, t2i);
  }
  for (int j = 0; j < 8; ++j) {
    int b = j + 8 * khalf;
    float sv = ws_s[b * 1024 + mx * 32 + my];
    long off = ((long)(b * 32 + o)) * 2048 + kxoff * 32 + my;
    oft_re[off] = t1r[j] + sv * t2r[j];
    oft_im[off] = t1i[j] + sv * t2i[j];
  }
}

// ---------------- Stage D: inverse column DFT (complex) ----------------
__global__ void __launch_bounds__(256) stageD_kernel(const float* __restrict__ oft_re,
                                                     const float* __restrict__ oft_im,
                                                     const float* __restrict__ ws,
                                                     float* __restrict__ ycol) {
  __shared__ float shCS[32768];               // cos|sin in [kx][h] layout, 128 KB
  lds_stage(shCS, ws + WS_CB, 32768);         // C_B then S_B (contiguous)
  const float* shC = shCS;
  const float* shS = shCS + 16384;
  int wid = (int)((blockIdx.x * 256 + threadIdx.x) >> 5);
  int img = wid >> 4;                 // 512 images (b,o)
  int mtile = wid & 15;               // h block
  int l = threadIdx.x & 31, lrow = l & 15, khalf = l >> 4;
  const float* P = oft_re + (long)img * 2048;
  const float* Q = oft_im + (long)img * 2048;
  int h = mtile * 16 + lrow;
  v8f cp0{}, cq0{}, sp0{}, sq0{}, cp1{}, cq1{}, sp1{}, sq1{};
  for (int k = 0; k < 64; k += 4) {
    int ka = k + 2 * khalf;           // kx index
    v2f ac; ac.x = shC[ka * 256 + h]; ac.y = shC[(ka + 1) * 256 + h];
    v2f as; as.x = shS[ka * 256 + h]; as.y = shS[(ka + 1) * 256 + h];
    v2f bp, bq;
    bp.x = P[ka * 32 + lrow];      bp.y = P[(ka + 1) * 32 + lrow];
    bq.x = Q[ka * 32 + lrow];      bq.y = Q[(ka + 1) * 32 + lrow];
    cp0 = wmma_f32(ac, bp, cp0);   cq0 = wmma_f32(ac, bq, cq0);
    sp0 = wmma_f32(as, bp, sp0);   sq0 = wmma_f32(as, bq, sq0);
    bp.x = P[ka * 32 + 16 + lrow]; bp.y = P[(ka + 1) * 32 + 16 + lrow];
    bq.x = Q[ka * 32 + 16 + lrow]; bq.y = Q[(ka + 1) * 32 + 16 + lrow];
    cp1 = wmma_f32(ac, bp, cp1);   cq1 = wmma_f32(ac, bq, cq1);
    sp1 = wmma_f32(as, bp, sp1);   sq1 = wmma_f32(as, bq, sq1);
  }
  float* Y = ycol + (long)img * 256 * 64;
  for (int j = 0; j < 8; ++j) {
    int hh = mtile * 16 + j + 8 * khalf;
    float* Yr = Y + (long)hh * 64;
    Yr[lrow]      = cp0[j] - sq0[j];   // Re = C*p - S*q
    Yr[16 + lrow] = cp1[j] - sq1[j];
    Yr[32 + lrow] = cq0[j] + sp0[j];   // Im = C*q + S*p
    Yr[48 + lrow] = cq1[j] + sp1[j];
  }
}

// ---------------- Stage E: inverse row rDFT  [131072x64] x [64x256] ----------------
__global__ void __launch_bounds__(256) stageE_kernel(const float* __restrict__ ycol,
                                                     const float* __restrict__ E,
                                                     float* __restrict__ out) {
  __shared__ float shE[16384];                // MatE [64][256], 64 KB
  lds_stage(shE, E, 16384);
  int wid = (int)((blockIdx.x * 256 + threadIdx.x) >> 5);
  int mblk = wid >> 2;
  int ng = wid & 3;
  int l = threadIdx.x & 31, lrow = l & 15, khalf = l >> 4;
  long row = (long)mblk * 16 + lrow;
  const float* Ar = ycol + row * 64;
  int n0 = ng * 64;
  v8f a0{}, a1{}, a2{}, a3{};
  for (int k = 0; k < 64; k += 4) {
    int ka = k + 2 * khalf;
    v2f a; a.x = Ar[ka]; a.y = Ar[ka + 1];
    const float* b0 = shE + ka * 256 + n0;
    const float* b1 = b0 + 256;
    v2f b;
    b.x = b0[lrow];      b.y = b1[lrow];      a0 = wmma_f32(a, b, a0);
    b.x = b0[16 + lrow]; b.y = b1[16 + lrow]; a1 = wmma_f32(a, b, a1);
    b.x = b0[32 + lrow]; b.y = b1[32 + lrow]; a2 = wmma_f32(a, b, a2);
    b.x = b0[48 + lrow]; b.y = b1[48 + lrow]; a3 = wmma_f32(a, b, a3);
  }
  long m0 = (long)mblk * 16;
  for (int j = 0; j < 8; ++j) {
    float* Dr = out + (m0 + j + 8 * khalf) * 256 + n0;
    Dr[lrow] = a0[j]; Dr[16 + lrow] = a1[j]; Dr[32 + lrow] = a2[j]; Dr[48 + lrow] = a3[j];
  }
}

// ---------------- launch ----------------
extern "C" void kernel_launch(void* const* d_in, const int* in_sizes, int n_in,
                              void* d_out, int out_size, void* d_ws, size_t ws_size,
                              hipStream_t stream) {
  const float* x   = (const float*)d_in[0];
  const float* fs  = (const float*)d_in[1];
  const float* w0  = (const float*)d_in[2];
  const float* w1  = (const float*)d_in[3];
  const float* we0 = (const float*)d_in[4];
  const float* we1 = (const float*)d_in[5];
  float* out = (float*)d_out;
  float* ws  = (float*)d_ws;

  float* xrow = ws + WS_XROW;
  float* xftr = ws + WS_XFTR;
  float* xfti = ws + WS_XFTI;
  float* oftr = ws + WS_OFTR;
  float* ofti = ws + WS_OFTI;
  float* ycol = xrow;  // Xrow dead after stage B -> reuse for Ycol

  precompute_kernel<<<(7 * 16384) / 256, 256, 0, stream>>>(ws, fs);
  stageA_kernel<<<1024, 256, 0, stream>>>(x, ws + WS_MATA, xrow);
  stageB_kernel<<<512, 256, 0, stream>>>(xrow, ws, xftr, xfti);
  stageC_kernel<<<512, 256, 0, stream>>>(xftr, xfti, w0, w1, we0, we1, ws + WS_S, oftr, ofti);
  stageD_kernel<<<1024, 256, 0, stream>>>(oftr, ofti, ws, ycol);
  stageE_kernel<<<4096, 256, 0, stream>>>(ycol, ws + WS_MATE, out);
}